// VSFE_35364760715982
// MI455X (gfx1250) — compile-verified
//
#include <hip/hip_runtime.h>
#include <hip/hip_bf16.h>

typedef float v2f __attribute__((ext_vector_type(2)));
typedef float v8f __attribute__((ext_vector_type(8)));

#define BATCH 256
#define CHN   16
#define SEQ   4096
#define KW    31
#define IMG   64
#define OHW   65
#define NBL   (BATCH * SEQ)   // elements per channel for BN stats
#define HALO  16              // LDS stage halo (16B-aligned interior for b128 async)

__device__ __forceinline__ float gelu_exact(float x) {
    return 0.5f * x * (1.0f + erff(x * 0.70710678118654752f));
}

// One 16-channel x 16-position conv1d tile via 8 chained V_WMMA_F32_16X16X4_F32.
// B operands come from a zero-padded LDS copy: sx[off + pos] == x[pos].
// A (16x4, M=channel, K=tap): lane idx = row M; regs hold K = {2*half, 2*half+1}.
// B (4x16, K=tap, N=position): lane idx = col N; regs hold K = {2*half, 2*half+1}.
__device__ __forceinline__ v8f conv_tile_lds(const float* __restrict__ sx, int k0,
                                             const v2f* a_step) {
    v8f acc = {};
#pragma unroll
    for (int s = 0; s < 8; ++s) {
        const int k = k0 + 4 * s;       // k0 = off + p0 + idx - 15 + koff
        v2f bb;
        bb.x = sx[k];
        bb.y = sx[k + 1];
        acc = __builtin_amdgcn_wmma_f32_16x16x4_f32(false, a_step[s], false, bb,
                                                    (short)0, acc, false, false);
    }
    return acc;
}

// Per-lane A operands: conv weights with channel_weights folded in.
__device__ __forceinline__ void build_a(const float* __restrict__ w1d,
                                        const float* __restrict__ cw,
                                        int idx, int koff, v2f* a_step) {
    const float cwv = cw[idx];
#pragma unroll
    for (int s = 0; s < 8; ++s) {
        const int t0 = 4 * s + koff;
        a_step[s].x = (t0     < KW) ? w1d[idx * KW + t0]     * cwv : 0.0f;
        a_step[s].y = (t0 + 1 < KW) ? w1d[idx * KW + t0 + 1] * cwv : 0.0f;
    }
}

// ---------------------------------------------------------------------------
// Pass 1: conv1d (WMMA f32) + bias + cw + GELU, accumulate per-channel sum/sq.
// One block per batch row. Row staged into LDS with the CDNA5 async copy path
// (GLOBAL_LOAD_ASYNC_TO_LDS_B128, ASYNCcnt); inner loop is ds_load + wmma only.
// ---------------------------------------------------------------------------
__global__ __launch_bounds__(256) void VSFE_conv_stats(
    const float* __restrict__ x, const float* __restrict__ w1d,
    const float* __restrict__ b1d, const float* __restrict__ cw,
    float* __restrict__ stats)
{
    __shared__ float sx[HALO + SEQ + HALO];   // sx[HALO + pos] = x[pos], zero halo

    const int lane = threadIdx.x & 31;
    const int wave = threadIdx.x >> 5;
    const int idx  = lane & 15;
    const int half = lane >> 4;
    const int koff = half * 2;
    const float* xr = x + (size_t)blockIdx.x * SEQ;

    // Async stage: 1024 float4 items, 4 per thread. GVS addressing:
    // mem = SADDR(xr) + VADDR(byte offset); VDST = per-lane LDS byte address.
#pragma unroll
    for (int q = 0; q < 4; ++q) {
        const int item = threadIdx.x + q * 256;                  // 0..1023
        const unsigned ldsaddr = (unsigned)(uintptr_t)(&sx[HALO + item * 4]);
        const unsigned goff    = (unsigned)(item * 16);          // bytes
        asm volatile("global_load_async_to_lds_b128 %0, %1, %2 offset:0"
                     :: "v"(ldsaddr), "v"(goff), "s"(xr)
                     : "memory");
    }
    if (threadIdx.x < HALO) {                 // zero halos (plain ds stores)
        sx[threadIdx.x] = 0.0f;
        sx[HALO + SEQ + threadIdx.x] = 0.0f;
    }
    asm volatile("s_wait_asynccnt 0x0" ::: "memory");
    __syncthreads();

    v2f a_step[8];
    build_a(w1d, cw, idx, koff, a_step);

    float biasv[8];
#pragma unroll
    for (int j = 0; j < 8; ++j) {
        const int ch = j + 8 * half;          // D layout: VGPR j -> channel j + 8*half
        biasv[j] = b1d[ch] * cw[ch];
    }

    float s1[8] = {0, 0, 0, 0, 0, 0, 0, 0};
    float s2[8] = {0, 0, 0, 0, 0, 0, 0, 0};
    const int kbase = HALO + idx - 15 + koff;
    for (int q = 0; q < 32; ++q) {
        const int p0 = (wave * 32 + q) * 16;
        v8f acc = conv_tile_lds(sx, kbase + p0, a_step);
#pragma unroll
        for (int j = 0; j < 8; ++j) {
            const float v = gelu_exact(acc[j] + biasv[j]);
            s1[j] += v;
            s2[j] += v * v;
        }
    }
    // reduce across the 16 lanes sharing a channel set (masks stay intra-half)
#pragma unroll
    for (int m = 8; m >= 1; m >>= 1) {
#pragma unroll
        for (int j = 0; j < 8; ++j) {
            s1[j] += __shfl_xor(s1[j], m, 32);
            s2[j] += __shfl_xor(s2[j], m, 32);
        }
    }
    if (idx == 0) {
#pragma unroll
        for (int j = 0; j < 8; ++j) {
            const int ch = j + 8 * half;
            atomicAdd(&stats[ch], s1[j]);
            atomicAdd(&stats[CHN + ch], s2[j]);
        }
    }
}

// ---------------------------------------------------------------------------
// Pass 2: recompute conv+GELU for image rows (i-1, i) from an LDS-staged x
// window, apply BN, stage x4 in LDS, fused effective 2x2 conv -> output row i.
// Block = (batch b, output row i). Waves: r = wave>>2 (row), t = wave&3 (tile).
// ---------------------------------------------------------------------------
__global__ __launch_bounds__(256) void VSFE_bn_conv2d(
    const float* __restrict__ x, const float* __restrict__ w1d,
    const float* __restrict__ b1d, const float* __restrict__ cw,
    const float* __restrict__ bnw, const float* __restrict__ bnb,
    const float* __restrict__ w_pw1, const float* __restrict__ b_pw1,
    const float* __restrict__ w_pw2, const float* __restrict__ b_pw2,
    const float* __restrict__ stats, float* __restrict__ out)
{
    __shared__ float lds[2][CHN][IMG];   // BN-normalized x4 rows (i-1, i)
    __shared__ float sx2[160];           // x window [(i-1)*64-15 .. i*64+79]

    const int lane = threadIdx.x & 31;
    const int wave = threadIdx.x >> 5;
    const int idx  = lane & 15;
    const int half = lane >> 4;
    const int koff = half * 2;
    const int b = blockIdx.x / OHW;
    const int i = blockIdx.x % OHW;
    const float* xr = x + (size_t)b * SEQ;

    const int r  = wave >> 2;       // local row 0/1
    const int t  = wave & 3;        // 16-position tile within the 64-wide row
    const int gr = i - 1 + r;       // global image row feeding output row i
    const int base = (i - 1) * IMG - 15;

    // Stage x window with zero padding at sequence edges (tiny: 160 floats).
    if (threadIdx.x < 160) {
        const int pos = base + threadIdx.x;
        sx2[threadIdx.x] = (pos >= 0 && pos < SEQ) ? xr[pos] : 0.0f;
    }

    v2f a_step[8];
    build_a(w1d, cw, idx, koff, a_step);

    float biasv[8], mu[8], sc[8], sh[8];
#pragma unroll
    for (int j = 0; j < 8; ++j) {
        const int ch = j + 8 * half;
        biasv[j] = b1d[ch] * cw[ch];
        const float mean = stats[ch] * (1.0f / (float)NBL);
        const float var  = stats[CHN + ch] * (1.0f / (float)NBL) - mean * mean;
        mu[j] = mean;
        sc[j] = rsqrtf(var + 1e-5f) * bnw[ch];
        sh[j] = bnb[ch];
    }
    __syncthreads();

    if (gr >= 0 && gr < IMG) {      // wave-uniform: EXEC all-ones at the WMMAs
        const int p0 = gr * IMG + t * 16;
        const int k0 = (p0 - base) + idx - 15 + koff;
        v8f acc = conv_tile_lds(sx2, k0, a_step);
#pragma unroll
        for (int j = 0; j < 8; ++j) {
            float v = gelu_exact(acc[j] + biasv[j]);
            v = (v - mu[j]) * sc[j] + sh[j];
            lds[r][j + 8 * half][t * 16 + idx] = v;
        }
    } else {
#pragma unroll
        for (int j = 0; j < 8; ++j)
            lds[r][j + 8 * half][t * 16 + idx] = 0.0f;
    }
    __syncthreads();

    // Conv2d(1->3,k=2,pad=1) + Conv2d(3->1,k=1) collapsed to one 2x2 stencil.
    float w00 = 0.f, w01 = 0.f, w10 = 0.f, w11 = 0.f, beff = b_pw2[0];
#pragma unroll
    for (int o = 0; o < 3; ++o) {
        const float wo = w_pw2[o];
        w00 += wo * w_pw1[o * 4 + 0];
        w01 += wo * w_pw1[o * 4 + 1];
        w10 += wo * w_pw1[o * 4 + 2];
        w11 += wo * w_pw1[o * 4 + 3];
        beff += wo * b_pw1[o];
    }

    for (int k = threadIdx.x; k < CHN * OHW; k += 256) {
        const int c  = k / OHW;
        const int j  = k % OHW;      // output column 0..64
        const int jm = j - 1;
        const float a00 = (jm >= 0)  ? lds[0][c][jm] : 0.0f;
        const float a01 = (j  < IMG) ? lds[0][c][j]  : 0.0f;
        const float a10 = (jm >= 0)  ? lds[1][c][jm] : 0.0f;
        const float a11 = (j  < IMG) ? lds[1][c][j]  : 0.0f;
        const float y = w00 * a00 + w01 * a01 + w10 * a10 + w11 * a11 + beff;
        const size_t oi = (((size_t)b * CHN + c) * OHW + i) * OHW + j;
        out[oi] = y * cw[c];
    }
}

extern "C" void kernel_launch(void* const* d_in, const int* in_sizes, int n_in,
                              void* d_out, int out_size, void* d_ws, size_t ws_size,
                              hipStream_t stream) {
    const float* x     = (const float*)d_in[0];
    const float* w1d   = (const float*)d_in[1];
    const float* b1d   = (const float*)d_in[2];
    const float* cw    = (const float*)d_in[3];
    const float* bnw   = (const float*)d_in[4];
    const float* bnb   = (const float*)d_in[5];
    const float* w_pw1 = (const float*)d_in[6];
    const float* b_pw1 = (const float*)d_in[7];
    const float* w_pw2 = (const float*)d_in[8];
    const float* b_pw2 = (const float*)d_in[9];
    float* stats = (float*)d_ws;            // [16 sums][16 sumsqs]
    float* out   = (float*)d_out;

    (void)hipMemsetAsync(stats, 0, 2 * CHN * sizeof(float), stream);  // deterministic
    VSFE_conv_stats<<<BATCH, 256, 0, stream>>>(x, w1d, b1d, cw, stats);
    VSFE_bn_conv2d<<<BATCH * OHW, 256, 0, stream>>>(x, w1d, b1d, cw, bnw, bnb,
                                                    w_pw1, b_pw1, w_pw2, b_pw2,
                                                    stats, out);
}